// DCNv2_24739011625366
// MI455X (gfx1250) — compile-verified
//
#include <hip/hip_runtime.h>
#include <hip/hip_bf16.h>
#include <math.h>

// ---------------------------------------------------------------------------
// DCNv2 for gfx1250 (MI455X), fp32 end-to-end using V_WMMA_F32_16X16X4_F32.
// B=8, H=W=64, C=256, F=256, K=3 (KK=9), stride=1, pad=1, dil=1.
// ---------------------------------------------------------------------------

typedef __attribute__((ext_vector_type(2))) float v2f;
typedef __attribute__((ext_vector_type(8))) float v8f;

#define BB    8
#define HH    64
#define WW    64
#define CC    256
#define FF    256
#define KKPTS 9
#define OMCH  27            // 18 offsets + 9 masks
#define NPIX  (BB * HH * WW)   // 32768
#define KTOT  (KKPTS * CC)     // 2304

// ---------------------------------------------------------------------------
// Kernel 1: offset/mask 3x3 conv. One wave32 per output pixel.
// Each lane strides over the 2304-element im2col patch, keeping 27 partial
// accumulators; wave-level xor-shuffle reduction; lane 0 writes raw om.
// ---------------------------------------------------------------------------
__global__ __launch_bounds__(256) void dcn_om_conv(
    const float* __restrict__ x,     // [B,H,W,C]
    const float* __restrict__ wk,    // [3,3,C,27] row-major -> [e][j], e=kk*256+c
    const float* __restrict__ wb,    // [27]
    float* __restrict__ om)          // [NPIX,27] raw (pre-sigmoid)
{
    const int gtid = blockIdx.x * blockDim.x + threadIdx.x;
    const int wave = gtid >> 5;
    const int lane = gtid & 31;
    if (wave >= NPIX) return;

    const int b   = wave >> 12;
    const int rem = wave & 4095;
    const int oy  = rem >> 6;
    const int ox  = rem & 63;

    float acc[OMCH];
#pragma unroll
    for (int j = 0; j < OMCH; ++j) acc[j] = 0.0f;

    for (int e = lane; e < KTOT; e += 32) {
        const int kk = e >> 8;        // kernel point 0..8
        const int c  = e & 255;
        const int ky = kk / 3;
        const int kx = kk - 3 * ky;
        const int iy = oy - 1 + ky;
        const int ix = ox - 1 + kx;
        float xv = 0.0f;
        if ((unsigned)iy < (unsigned)HH && (unsigned)ix < (unsigned)WW)
            xv = x[(((b * HH) + iy) * WW + ix) * CC + c];
        const float* wrow = wk + e * OMCH;
#pragma unroll
        for (int j = 0; j < OMCH; ++j) acc[j] = fmaf(xv, wrow[j], acc[j]);
    }

    // wave32 reduction
#pragma unroll
    for (int off = 16; off > 0; off >>= 1) {
#pragma unroll
        for (int j = 0; j < OMCH; ++j)
            acc[j] += __shfl_xor(acc[j], off, 32);
    }

    if (lane == 0) {
        float* op = om + wave * OMCH;
#pragma unroll
        for (int j = 0; j < OMCH; ++j) op[j] = acc[j] + wb[j];
    }
}

// ---------------------------------------------------------------------------
// Kernel 2: fused deformable sampling + GEMM.
// Block = 512 threads = 16 waves. Block owns 16 pixels (M-tile); wave w owns
// output channels [16w, 16w+16) (N-slab). For each of the 9 kernel points:
//   phase 1: build cols slice [16 px][256 ch] in LDS (bilinear * mask)
//   phase 2: 64 x v_wmma_f32_16x16x4_f32 per wave, A from LDS, B from L2.
// Accumulators live in v8f across all 576 WMMAs; epilogue adds bias.
// ---------------------------------------------------------------------------
#define LDS_STRIDE 260   // 256 + 4 pad: 260 % 64 == 4 -> conflict-free A reads

__global__ __launch_bounds__(512) void dcn_fused_gemm(
    const float* __restrict__ x,     // [B,H,W,C]
    const float* __restrict__ om,    // [NPIX,27] raw
    const float* __restrict__ wgt,   // [2304,256]
    const float* __restrict__ bias,  // [256]
    float* __restrict__ out)         // [NPIX,256]
{
    __shared__ float cols[16][LDS_STRIDE];

    const int tid  = threadIdx.x;
    const int lane = tid & 31;
    const int wid  = tid >> 5;            // 0..15
    const int p0   = blockIdx.x * 16;     // first pixel of M-tile

    // phase-1 identity: wave `wid` samples pixel p0+wid
    const int p   = p0 + wid;
    const int b   = p >> 12;
    const int rem = p & 4095;
    const int oy  = rem >> 6;
    const int ox  = rem & 63;
    const float* omp = om + p * OMCH;

    // phase-2 identity
    const int n0     = wid * 16;          // N-slab base
    const int m_lane = lane & 15;         // A row / D column within slab
    const int kh     = lane >> 4;         // K-half selector (ISA fp32 layouts)

    v8f acc = {0.f, 0.f, 0.f, 0.f, 0.f, 0.f, 0.f, 0.f};

    for (int kk = 0; kk < KKPTS; ++kk) {
        // ---------------- phase 1: deformable bilinear sample ----------------
        const float dy = omp[2 * kk + 0];
        const float dx = omp[2 * kk + 1];
        const float mr = omp[18 + kk];
        const float msk = 2.0f / (1.0f + __expf(-mr));   // sigmoid * 2

        const int ky = kk / 3;
        const int kx = kk - 3 * ky;
        const float py = (float)(oy - 1 + ky) + dy;
        const float px = (float)(ox - 1 + kx) + dx;
        const float fy0 = floorf(py), fx0 = floorf(px);
        const float wy1 = py - fy0, wy0 = 1.0f - wy1;
        const float wx1 = px - fx0, wx0 = 1.0f - wx1;
        const int y0 = (int)fy0, x0 = (int)fx0;
        const int y1 = y0 + 1,  x1 = x0 + 1;

        const bool vy0 = (unsigned)y0 < (unsigned)HH;
        const bool vy1 = (unsigned)y1 < (unsigned)HH;
        const bool vx0 = (unsigned)x0 < (unsigned)WW;
        const bool vx1 = (unsigned)x1 < (unsigned)WW;

        // fold validity + modulation mask into corner weights
        const float c00 = (vy0 && vx0) ? wy0 * wx0 * msk : 0.0f;
        const float c01 = (vy0 && vx1) ? wy0 * wx1 * msk : 0.0f;
        const float c10 = (vy1 && vx0) ? wy1 * wx0 * msk : 0.0f;
        const float c11 = (vy1 && vx1) ? wy1 * wx1 * msk : 0.0f;

        const int yc0 = min(max(y0, 0), HH - 1);
        const int yc1 = min(max(y1, 0), HH - 1);
        const int xc0 = min(max(x0, 0), WW - 1);
        const int xc1 = min(max(x1, 0), WW - 1);
        const long i00 = (long)(((b * HH) + yc0) * WW + xc0) * CC;
        const long i01 = (long)(((b * HH) + yc0) * WW + xc1) * CC;
        const long i10 = (long)(((b * HH) + yc1) * WW + xc0) * CC;
        const long i11 = (long)(((b * HH) + yc1) * WW + xc1) * CC;

#pragma unroll
        for (int cb = 0; cb < CC; cb += 32) {
            const int c = cb + lane;
            const float v = c00 * x[i00 + c] + c01 * x[i01 + c] +
                            c10 * x[i10 + c] + c11 * x[i11 + c];
            cols[wid][c] = v;
        }
        __syncthreads();

        // ---------------- phase 2: WMMA over this 256-deep K slice ----------
        // ISA fp32 A 16x4 layout: lanes 0-15 hold {K+0,K+1}, lanes 16-31 hold
        // {K+2,K+3}; B 4x16 mirrored: VGPR0 = rows K+2*kh, VGPR1 = rows +1.
        const float* wb_ptr = wgt + (long)(kk * CC) * FF + n0 + m_lane;
        for (int k = 0; k < CC; k += 4) {
            const v2f a = *(const v2f*)&cols[m_lane][k + 2 * kh];
            v2f bf;
            bf.x = wb_ptr[(long)(k + 2 * kh + 0) * FF];
            bf.y = wb_ptr[(long)(k + 2 * kh + 1) * FF];
            acc = __builtin_amdgcn_wmma_f32_16x16x4_f32(
                false, a, false, bf, (short)0, acc, false, false);
        }
        __syncthreads();
    }

    // ---------------- epilogue: bias + store --------------------------------
    // C/D layout: VGPR j -> M = j (lanes 0-15) / j+8 (lanes 16-31), N = lane%16
    const int n = n0 + m_lane;
    const float bv = bias[n];
    const int mbase = kh * 8;
#pragma unroll
    for (int j = 0; j < 8; ++j) {
        out[(long)(p0 + mbase + j) * FF + n] = acc[j] + bv;
    }
}

// ---------------------------------------------------------------------------
// Launch: inputs in setup_inputs() order:
//   d_in[0] inputs   (8,64,64,256) f32
//   d_in[1] om_kernel(3,3,256,27)  f32
//   d_in[2] om_bias  (27)          f32
//   d_in[3] kernel   (2304,256)    f32
//   d_in[4] bias     (1,1,1,256)   f32
// d_out: (8,64,64,256) f32.  d_ws: om scratch [32768*27] f32 (3.5 MB).
// ---------------------------------------------------------------------------
extern "C" void kernel_launch(void* const* d_in, const int* in_sizes, int n_in,
                              void* d_out, int out_size, void* d_ws, size_t ws_size,
                              hipStream_t stream) {
    const float* x    = (const float*)d_in[0];
    const float* omk  = (const float*)d_in[1];
    const float* omb  = (const float*)d_in[2];
    const float* wgt  = (const float*)d_in[3];
    const float* bias = (const float*)d_in[4];
    float* out = (float*)d_out;
    float* om  = (float*)d_ws;   // 32768 * 27 floats

    // Kernel 1: 32768 waves (one per pixel), 8 waves per block.
    dcn_om_conv<<<NPIX / 8, 256, 0, stream>>>(x, omk, omb, om);

    // Kernel 2: 2048 M-tiles of 16 pixels, 16 waves per block.
    dcn_fused_gemm<<<NPIX / 16, 512, 0, stream>>>(x, om, wgt, bias, out);
}